// STSEAGWNN_71055938945724
// MI455X (gfx1250) — compile-verified
//
#include <hip/hip_runtime.h>
#include <hip/hip_bf16.h>
#include <math.h>

#define B_   8
#define T_   12
#define N_   207
#define C_   32
#define SK_  256
#define E1_  512
#define OUT_ 12
#define NL_  8
#define NP_  208
#define M_   (B_*T_*N_)   // 19872

typedef float v2f __attribute__((ext_vector_type(2)));
typedef float v8f __attribute__((ext_vector_type(8)));

__device__ inline v8f zero8() {
  v8f z;
  #pragma unroll
  for (int i = 0; i < 8; ++i) z[i] = 0.f;
  return z;
}

__device__ inline v8f wmma4(v2f a, v2f b, v8f c) {
  // D = A(16x4,f32) * B(4x16,f32) + C(16x16,f32)
  return __builtin_amdgcn_wmma_f32_16x16x4_f32(false, a, false, b, (short)0, c, false, false);
}

__device__ inline float sigmoidf_(float x) { return 1.f / (1.f + __expf(-x)); }
__device__ inline float leakyf_(float x)   { return x > 0.f ? x : 0.2f * x; }

// ---------------- start: h[b,t,n,c] = start_w[c]*x[b,t,n] + start_b[c] ----------------
__global__ __launch_bounds__(256) void k_start(const float* __restrict__ x,
                                               const float* __restrict__ sw,
                                               const float* __restrict__ sb,
                                               float* __restrict__ h) {
  int i = blockIdx.x * 256 + threadIdx.x;
  if (i >= B_ * T_ * N_ * C_) return;
  int c = i & 31, p = i >> 5;
  h[i] = sw[c] * x[p] + sb[c];
}

// ---------------- time attention A_time (B,12,12) ----------------
__global__ __launch_bounds__(128) void k_time_attn(const float* __restrict__ h,
                                                   const float* __restrict__ wqw,
                                                   const float* __restrict__ wqb,
                                                   const float* __restrict__ wkw,
                                                   const float* __restrict__ wkb,
                                                   float* __restrict__ At) {
  __shared__ float Xt[12][32], Q[12][32], Kk[12][32], sc[12][12];
  int b = blockIdx.x, tid = threadIdx.x;
  for (int i = tid; i < 12 * 32; i += 128) {
    int t = i >> 5, c = i & 31;
    float s = 0.f;
    const float* hp = h + ((size_t)(b * T_ + t) * N_) * C_ + c;
    for (int n = 0; n < N_; ++n) s += hp[n * C_];
    Xt[t][c] = s / (float)N_;
  }
  __syncthreads();
  for (int i = tid; i < 12 * 32; i += 128) {
    int t = i >> 5, c = i & 31;
    float q = 0.f, k = 0.f;
    for (int kk = 0; kk < 32; ++kk) { q += Xt[t][kk] * wqw[c * 32 + kk]; k += Xt[t][kk] * wkw[c * 32 + kk]; }
    Q[t][c] = q + wqb[c]; Kk[t][c] = k + wkb[c];
  }
  __syncthreads();
  for (int i = tid; i < 144; i += 128) {
    int r = i / 12, cl = i % 12;
    float s = 0.f;
    for (int c = 0; c < 32; ++c) s += Q[r][c] * Kk[cl][c];
    sc[r][cl] = s * 0.17677669529663687f; // 1/sqrt(32)
  }
  __syncthreads();
  if (tid < 12) {
    int r = tid;
    float mx = -3.4e38f;
    for (int j = 0; j < 12; ++j) mx = fmaxf(mx, sc[r][j]);
    float e[12], sum = 0.f;
    for (int j = 0; j < 12; ++j) { e[j] = __expf(sc[r][j] - mx); sum += e[j]; }
    for (int j = 0; j < 12; ++j) {
      float v = e[j] / sum;
      if (j == r) v = 0.f;
      At[(b * 12 + r) * 12 + j] = v;
    }
  }
}

// ---------------- skip GEMM: skip[p,o] (+)= sum_c h[p,c]*W[o,c] + bias[o] ----------------
__global__ __launch_bounds__(256) void k_skip(const float* __restrict__ h,
                                              const float* __restrict__ W,
                                              const float* __restrict__ bias,
                                              float* __restrict__ skip, int first) {
  int mt = blockIdx.x;
  int lane = threadIdx.x & 31, wv = threadIdx.x >> 5;
  int half = lane >> 4, rc = lane & 15;
  const float* Arow = h + (size_t)(mt * 16 + rc) * C_;
  for (int nt = wv * 2; nt < wv * 2 + 2; ++nt) {
    int o = nt * 16 + rc;
    const float* Brow = W + (size_t)o * C_;
    v8f acc = zero8();
    for (int kt = 0; kt < 2; ++kt) {
      #pragma unroll
      for (int s = 0; s < 4; ++s) {
        int kb = kt * 16 + 4 * s + 2 * half;
        v2f a;  a.x = Arow[kb]; a.y = Arow[kb + 1];
        v2f bb; bb.x = Brow[kb]; bb.y = Brow[kb + 1];
        acc = wmma4(a, bb, acc);
      }
    }
    float bo = bias[o];
    #pragma unroll
    for (int v = 0; v < 8; ++v) {
      int p = mt * 16 + v + 8 * half;
      float val = acc[v] + bo;
      float* dst = &skip[(size_t)p * SK_ + o];
      if (first) *dst = val; else *dst += val;
    }
  }
}

// ---------------- spatial attention: build U (96,208,208) + rowsum ----------------
__global__ __launch_bounds__(256) void k_sp_attn(const float* __restrict__ h,
                                                 const float* __restrict__ adj,
                                                 const float* __restrict__ a_sp,
                                                 float* __restrict__ U,
                                                 float* __restrict__ rowsum) {
  __shared__ float f1[N_], f2[N_];
  int bt = blockIdx.x, tid = threadIdx.x;
  const float* X = h + (size_t)bt * N_ * C_;
  for (int n = tid; n < N_; n += 256) {
    float s1 = 0.f, s2 = 0.f;
    for (int c = 0; c < 32; ++c) { float xv = X[n * 32 + c]; s1 += xv * a_sp[c]; s2 += xv * a_sp[32 + c]; }
    f1[n] = s1; f2[n] = s2;
  }
  __syncthreads();
  float* Ub = U + (size_t)bt * NP_ * NP_;
  int lane = tid & 31, wv = tid >> 5;
  for (int r = wv; r < N_; r += 8) {
    float fr = f1[r];
    const float* arow = adj + (size_t)r * N_;
    float mx = -3.4e38f;
    for (int j = lane; j < N_; j += 32) {
      float v = (arow[j] > 0.f) ? leakyf_(fr + f2[j]) : -3.4e38f;
      mx = fmaxf(mx, v);
    }
    for (int m = 16; m >= 1; m >>= 1) mx = fmaxf(mx, __shfl_xor(mx, m, 32));
    float sum = 0.f;
    for (int j = lane; j < N_; j += 32) {
      float v = (arow[j] > 0.f) ? leakyf_(fr + f2[j]) : -3.4e38f;
      float e = __expf(v - mx);
      Ub[(size_t)r * NP_ + j] = e;
      sum += e;
    }
    for (int m = 16; m >= 1; m >>= 1) sum += __shfl_xor(sum, m, 32);
    float inv = 1.f / sum, rs = 0.f;
    for (int j = lane; j < N_; j += 32) {
      float u = Ub[(size_t)r * NP_ + j] * inv;
      Ub[(size_t)r * NP_ + j] = u;
      rs += arow[j] * 0.5f * u;
    }
    for (int m = 16; m >= 1; m >>= 1) rs += __shfl_xor(rs, m, 32);
    if (lane == 0) { rowsum[bt * NP_ + r] = rs; Ub[(size_t)r * NP_ + N_] = 0.f; }
  }
  for (int j = tid; j < NP_; j += 256) Ub[(size_t)N_ * NP_ + j] = 0.f; // pad row
}

// ---------------- spatial hops: per (b,t), 4x [U@upd + combine], WMMA f32 16x16x4 ----------------
__global__ __launch_bounds__(256) void k_sp_hops(const float* __restrict__ hin,
                                                 float* __restrict__ hout,
                                                 const float* __restrict__ U,
                                                 const float* __restrict__ rowsum,
                                                 const float* __restrict__ temp,
                                                 const float* __restrict__ cheb) {
  __shared__ float bufA[NP_ * C_], bufB[NP_ * C_];
  int bt = blockIdx.x, tid = threadIdx.x;
  const float* X = hin + (size_t)bt * N_ * C_;
  const float* Ub = U + (size_t)bt * NP_ * NP_;
  float coe0 = sigmoidf_(temp[0]), coe1 = sigmoidf_(temp[1]), coe2 = sigmoidf_(temp[2]);
  float xR[26], primeR[26], rsR[26];
  #pragma unroll
  for (int e = 0; e < 26; ++e) {
    int idx = tid + e * 256;
    if (idx < N_ * C_) {
      float xv = X[idx];
      xR[e] = xv; primeR[e] = xv;
      rsR[e] = rowsum[bt * NP_ + (idx >> 5)];
      bufA[idx] = xv;
    }
  }
  for (int idx = N_ * C_ + tid; idx < NP_ * C_; idx += 256) { bufA[idx] = 0.f; bufB[idx] = 0.f; }
  __syncthreads();
  float* updP = bufA; float* newP = bufB;
  int lane = tid & 31, wv = tid >> 5;
  int half = lane >> 4, rc = lane & 15;
  float chebf = 1.f;
  for (int hop = 0; hop < 4; ++hop) {
    for (int tile = wv; tile < 26; tile += 8) {          // wave-uniform trip count
      int rt = tile >> 1, ct = tile & 1;
      const float* Urow = Ub + (size_t)(rt * 16 + rc) * NP_;
      int col = ct * 16 + rc;
      v8f acc = zero8();
      for (int kt = 0; kt < 13; ++kt) {
        #pragma unroll
        for (int s = 0; s < 4; ++s) {
          int kb = kt * 16 + 4 * s + 2 * half;
          v2f a;  a.x = Urow[kb];             a.y = Urow[kb + 1];
          v2f bb; bb.x = updP[kb * 32 + col]; bb.y = updP[(kb + 1) * 32 + col];
          acc = wmma4(a, bb, acc);
        }
      }
      #pragma unroll
      for (int v = 0; v < 8; ++v)
        newP[(rt * 16 + v + 8 * half) * 32 + col] = acc[v]; // pad row gets 0 (U pad row is 0)
    }
    __syncthreads();
    if (hop >= 1) chebf *= sigmoidf_(cheb[hop - 1]);
    #pragma unroll
    for (int e = 0; e < 26; ++e) {
      int idx = tid + e * 256;
      if (idx < N_ * C_) {
        float u = newP[idx];
        float even = coe0 * xR[e] + u;
        float odd  = u - even * (rsR[e] * chebf);
        float fuse = coe1 * even + (1.f - coe1) * odd;
        float base = (hop == 0) ? xR[e] : primeR[e];
        primeR[e] = coe2 * base + (1.f - coe2) * fuse;
      }
    }
    float* t = updP; updP = newP; newP = t;
    __syncthreads();
  }
  #pragma unroll
  for (int e = 0; e < 26; ++e) {
    int idx = tid + e * 256;
    if (idx < N_ * C_) hout[(size_t)bt * N_ * C_ + idx] = primeR[e];
  }
}

// ---------------- temporal wavelet: one wave per (b,n); 12x12 padded to 16; + residual ----------------
__global__ __launch_bounds__(128) void k_tp(float* __restrict__ h,
                                            const float* __restrict__ hA,
                                            const float* __restrict__ At,
                                            const float* __restrict__ a_tp,
                                            const float* __restrict__ temp,
                                            const float* __restrict__ cheb) {
  __shared__ float Us[4][16 * 16];
  __shared__ float ua[4][16 * 32], ub[4][16 * 32];
  __shared__ float ff[4][48];
  int lane = threadIdx.x & 31, wv = threadIdx.x >> 5;
  int bn = blockIdx.x * 4 + wv;             // 414*4 == 1656 exactly
  int b = bn / N_, n = bn % N_;
  float* Uw = Us[wv];
  float* updP = ua[wv]; float* newP = ub[wv];
  float* f1 = ff[wv]; float* f2 = ff[wv] + 16; float* rsv = ff[wv] + 32;
  const float* Xb = hA + ((size_t)b * T_ * N_ + n) * C_;
  float coe0 = sigmoidf_(temp[0]), coe1 = sigmoidf_(temp[1]), coe2 = sigmoidf_(temp[2]);
  float xR[12], primeR[12];
  #pragma unroll
  for (int t = 0; t < 12; ++t) {
    float xv = Xb[(size_t)t * N_ * C_ + lane];
    xR[t] = xv; primeR[t] = xv;
    updP[t * 32 + lane] = xv;
  }
  #pragma unroll
  for (int t = 12; t < 16; ++t) { updP[t * 32 + lane] = 0.f; newP[t * 32 + lane] = 0.f; }
  __syncthreads();
  if (lane < 12) {
    float s1 = 0.f, s2 = 0.f;
    for (int c = 0; c < 32; ++c) { float xv = updP[lane * 32 + c]; s1 += xv * a_tp[c]; s2 += xv * a_tp[32 + c]; }
    f1[lane] = s1; f2[lane] = s2;
  }
  __syncthreads();
  if (lane < 16) {
    int r = lane;
    if (r < 12) {
      const float* arow = At + (b * 12 + r) * 12;
      float fr = f1[r], vb[12], mx = -3.4e38f;
      for (int j = 0; j < 12; ++j) {
        float v = (arow[j] > 0.f) ? leakyf_(fr + f2[j]) : -3.4e38f;
        vb[j] = v; mx = fmaxf(mx, v);
      }
      float sum = 0.f;
      for (int j = 0; j < 12; ++j) { vb[j] = __expf(vb[j] - mx); sum += vb[j]; }
      float inv = 1.f / sum, rs = 0.f;
      for (int j = 0; j < 12; ++j) { float u = vb[j] * inv; Uw[r * 16 + j] = u; rs += arow[j] * 0.5f * u; }
      for (int j = 12; j < 16; ++j) Uw[r * 16 + j] = 0.f;
      rsv[r] = rs;
    } else {
      for (int j = 0; j < 16; ++j) Uw[r * 16 + j] = 0.f;
      rsv[r] = 0.f;
    }
  }
  __syncthreads();
  int half = lane >> 4, rc = lane & 15;
  float chebf = 1.f;
  for (int hop = 0; hop < 4; ++hop) {
    #pragma unroll
    for (int ct = 0; ct < 2; ++ct) {
      v8f acc = zero8();
      #pragma unroll
      for (int s = 0; s < 4; ++s) {
        int kb = 4 * s + 2 * half;
        v2f a;  a.x = Uw[rc * 16 + kb];               a.y = Uw[rc * 16 + kb + 1];
        v2f bb; bb.x = updP[kb * 32 + ct * 16 + rc];  bb.y = updP[(kb + 1) * 32 + ct * 16 + rc];
        acc = wmma4(a, bb, acc);
      }
      #pragma unroll
      for (int v = 0; v < 8; ++v) newP[(v + 8 * half) * 32 + ct * 16 + rc] = acc[v];
    }
    __syncthreads();
    if (hop >= 1) chebf *= sigmoidf_(cheb[hop - 1]);
    #pragma unroll
    for (int t = 0; t < 12; ++t) {
      float u = newP[t * 32 + lane];
      float even = coe0 * xR[t] + u;
      float odd  = u - even * (rsv[t] * chebf);
      float fuse = coe1 * even + (1.f - coe1) * odd;
      float base = (hop == 0) ? xR[t] : primeR[t];
      primeR[t] = coe2 * base + (1.f - coe2) * fuse;
    }
    float* tp = updP; updP = newP; newP = tp;
    __syncthreads();
  }
  float* Hb = h + ((size_t)b * T_ * N_ + n) * C_;
  #pragma unroll
  for (int t = 0; t < 12; ++t) {
    size_t off = (size_t)t * N_ * C_ + lane;
    Hb[off] = primeR[t] + Hb[off];   // + residual, in place (wave-exclusive locations)
  }
}

// ---------------- per-channel batchnorm over (B,N,T) ----------------
__global__ __launch_bounds__(256) void k_bn(float* __restrict__ h) {
  __shared__ float r0[256], r1[256];
  int c = blockIdx.x, tid = threadIdx.x;
  float s = 0.f, s2 = 0.f;
  for (int p = tid; p < M_; p += 256) { float v = h[(size_t)p * 32 + c]; s += v; s2 += v * v; }
  r0[tid] = s; r1[tid] = s2;
  __syncthreads();
  for (int w = 128; w >= 1; w >>= 1) {
    if (tid < w) { r0[tid] += r0[tid + w]; r1[tid] += r1[tid + w]; }
    __syncthreads();
  }
  float mean = r0[0] / (float)M_;
  float var  = r1[0] / (float)M_ - mean * mean;
  float inv  = rsqrtf(var + 1e-5f);
  for (int p = tid; p < M_; p += 256) h[(size_t)p * 32 + c] = (h[(size_t)p * 32 + c] - mean) * inv;
}

// ---------------- fused end: relu(skip) -> relu(end1) -> end2, scatter to (B,12,N,T) ----------------
__global__ __launch_bounds__(256) void k_end(const float* __restrict__ skip,
                                             const float* __restrict__ w1, const float* __restrict__ b1,
                                             const float* __restrict__ w2, const float* __restrict__ b2,
                                             float* __restrict__ out) {
  __shared__ float sk[16 * SK_];
  __shared__ float t1[16 * E1_];
  int mt = blockIdx.x, tid = threadIdx.x;
  for (int i = tid; i < 16 * SK_; i += 256) {
    float v = skip[(size_t)mt * 16 * SK_ + i];
    sk[i] = v > 0.f ? v : 0.f;
  }
  __syncthreads();
  int lane = tid & 31, wv = tid >> 5, half = lane >> 4, rc = lane & 15;
  for (int ct = wv; ct < 32; ct += 8) {
    int o = ct * 16 + rc;
    const float* Brow = w1 + (size_t)o * SK_;
    v8f acc = zero8();
    for (int kt = 0; kt < 16; ++kt) {
      #pragma unroll
      for (int s = 0; s < 4; ++s) {
        int kb = kt * 16 + 4 * s + 2 * half;
        v2f a;  a.x = sk[rc * SK_ + kb]; a.y = sk[rc * SK_ + kb + 1];
        v2f bb; bb.x = Brow[kb];         bb.y = Brow[kb + 1];
        acc = wmma4(a, bb, acc);
      }
    }
    float bo = b1[o];
    #pragma unroll
    for (int v = 0; v < 8; ++v) {
      float val = acc[v] + bo;
      t1[(v + 8 * half) * E1_ + o] = val > 0.f ? val : 0.f;
    }
  }
  __syncthreads();
  if (wv == 0) {                                // wave-uniform branch, EXEC full inside
    int o = rc;
    float msk = (o < OUT_) ? 1.f : 0.f;
    const float* w2row = w2 + (size_t)(o < OUT_ ? o : 0) * E1_;
    v8f acc = zero8();
    for (int kt = 0; kt < 32; ++kt) {
      #pragma unroll
      for (int s = 0; s < 4; ++s) {
        int kb = kt * 16 + 4 * s + 2 * half;
        v2f a;  a.x = t1[rc * E1_ + kb]; a.y = t1[rc * E1_ + kb + 1];
        v2f bb; bb.x = w2row[kb] * msk;  bb.y = w2row[kb + 1] * msk;
        acc = wmma4(a, bb, acc);
      }
    }
    #pragma unroll
    for (int v = 0; v < 8; ++v) {
      if (o < OUT_) {
        int p = mt * 16 + v + 8 * half;
        int b = p / (T_ * N_), rem = p % (T_ * N_);
        int t = rem / N_, n = rem % N_;
        out[(((size_t)b * OUT_ + o) * N_ + n) * T_ + t] = acc[v] + b2[o];
      }
    }
  }
}

extern "C" void kernel_launch(void* const* d_in, const int* in_sizes, int n_in,
                              void* d_out, int out_size, void* d_ws, size_t ws_size,
                              hipStream_t stream) {
  const float* x       = (const float*)d_in[0];
  const float* adj     = (const float*)d_in[1];
  const float* start_w = (const float*)d_in[2];
  const float* start_b = (const float*)d_in[3];
  const float* wq_w    = (const float*)d_in[4];
  const float* wq_b    = (const float*)d_in[5];
  const float* wk_w    = (const float*)d_in[6];
  const float* wk_b    = (const float*)d_in[7];
  const float* skip_w  = (const float*)d_in[8];
  const float* skip_b  = (const float*)d_in[9];
  const float* a_sp    = (const float*)d_in[10];
  const float* temp_sp = (const float*)d_in[11];
  const float* cheb_sp = (const float*)d_in[12];
  const float* a_tp    = (const float*)d_in[13];
  const float* temp_tp = (const float*)d_in[14];
  const float* cheb_tp = (const float*)d_in[15];
  const float* end1_w  = (const float*)d_in[16];
  const float* end1_b  = (const float*)d_in[17];
  const float* end2_w  = (const float*)d_in[18];
  const float* end2_b  = (const float*)d_in[19];
  float* out = (float*)d_out;

  float* ws   = (float*)d_ws;
  float* h    = ws;                                  // 635904
  float* hA   = h   + (size_t)M_ * C_;               // 635904
  float* U    = hA  + (size_t)M_ * C_;               // 96*208*208
  float* rs   = U   + (size_t)B_ * T_ * NP_ * NP_;   // 96*208
  float* At   = rs  + (size_t)B_ * T_ * NP_;         // 1152
  float* skip = At  + 1280;                          // 19872*256

  k_start<<<(B_ * T_ * N_ * C_ + 255) / 256, 256, 0, stream>>>(x, start_w, start_b, h);
  k_time_attn<<<B_, 128, 0, stream>>>(h, wq_w, wq_b, wk_w, wk_b, At);

  for (int i = 0; i < NL_; ++i) {
    k_skip<<<M_ / 16, 256, 0, stream>>>(h, skip_w + (size_t)i * SK_ * C_, skip_b + i * SK_, skip, i == 0);
    if (i == NL_ - 1) break;
    k_sp_attn<<<B_ * T_, 256, 0, stream>>>(h, adj, a_sp + i * 2 * C_, U, rs);
    k_sp_hops<<<B_ * T_, 256, 0, stream>>>(h, hA, U, rs, temp_sp + i * 5, cheb_sp + i * 5);
    k_tp<<<(B_ * N_) / 4, 128, 0, stream>>>(h, hA, At, a_tp + i * 2 * C_, temp_tp + i * 5, cheb_tp + i * 5);
    k_bn<<<C_, 256, 0, stream>>>(h);
  }
  k_end<<<M_ / 16, 256, 0, stream>>>(skip, end1_w, end1_b, end2_w, end2_b, out);
}